// GrayscaleDilation2D_51797305589785
// MI455X (gfx1250) — compile-verified
//
#include <hip/hip_runtime.h>

#define KX 7
#define PAD 3
#define TILE 64
#define HALO (TILE + KX - 1)   /* 70 */
#define LDSW 72                /* padded LDS row stride (floats), keeps 16B alignment */
#define WIDTH 512
#define PLANES 128             /* 8 * 16 */
#define NEGINF (-1.0e30f)

#if __has_builtin(__builtin_amdgcn_global_load_async_to_lds_b32)
#define HAVE_ASYNC_LDS 1
#else
#define HAVE_ASYNC_LDS 0
#endif

#if __has_builtin(__builtin_amdgcn_s_wait_asynccnt)
#define WAIT_ASYNC() __builtin_amdgcn_s_wait_asynccnt(0)
#else
#define WAIT_ASYNC() asm volatile("s_wait_asynccnt 0" ::: "memory")
#endif

typedef __attribute__((address_space(1))) int* as1_ip;
typedef __attribute__((address_space(3))) int* as3_ip;

__global__ __launch_bounds__(256) void dilate7x7_kernel(
    const float* __restrict__ img,
    const float* __restrict__ filt,
    float* __restrict__ out) {
  __shared__ float tile[HALO * LDSW];
  __shared__ float sf[KX * KX];

  const int tid = threadIdx.x;
  const int p   = blockIdx.z;
  const int bx  = blockIdx.x * TILE;
  const int by  = blockIdx.y * TILE;
  const float* plane = img + (size_t)p * WIDTH * WIDTH;

  if (tid < KX * KX) sf[tid] = filt[tid];

  // ---- Stage 70x70 halo tile into LDS -------------------------------------
  // In-image cells: async global->LDS copy (ASYNCcnt).  Out-of-image cells:
  // plain ds_store of -1e30.  Addresses are disjoint, so the two engines
  // cannot race on the same cell.
  for (int idx = tid; idx < HALO * HALO; idx += 256) {
    const int ly  = idx / HALO;
    const int lx  = idx - ly * HALO;
    const int gy  = by - PAD + ly;
    const int gx  = bx - PAD + lx;
    const int off = ly * LDSW + lx;
    const bool in = (gx >= 0) & (gx < WIDTH) & (gy >= 0) & (gy < WIDTH);
#if HAVE_ASYNC_LDS
    if (in) {
      __builtin_amdgcn_global_load_async_to_lds_b32(
          (as1_ip)(const_cast<float*>(plane) + (size_t)gy * WIDTH + gx),
          (as3_ip)(&tile[off]), 0, 0);
    } else {
      tile[off] = NEGINF;
    }
#else
    tile[off] = in ? plane[(size_t)gy * WIDTH + gx] : NEGINF;
#endif
  }
#if HAVE_ASYNC_LDS
  WAIT_ASYNC();
#endif
  __syncthreads();

  // ---- Hoist the (uniform) 7x7 filter into registers ----------------------
  float fv[KX * KX];
#pragma unroll
  for (int k = 0; k < KX * KX; ++k) fv[k] = sf[k];

  // ---- Compute 4x4 outputs per thread, sliding over 10 input rows ---------
  // Each of the 10 LDS rows this thread needs is read exactly once (3 ds
  // loads: b128 + b128 + b64) and broadcast into all overlapping output-row
  // accumulators.
  const int ox0 = (tid & 15) * 4;   // 16 threads across x
  const int oy0 = (tid >> 4) * 4;   // 16 threads across y
  float* plane_out = out + (size_t)p * WIDTH * WIDTH;

  float acc[4][4];
#pragma unroll
  for (int oy = 0; oy < 4; ++oy)
#pragma unroll
    for (int x = 0; x < 4; ++x) acc[oy][x] = NEGINF;

  // Pairwise max trees -> v_max3_num_f32 candidates.
#define APPLY_ROW(A, B)                                          \
  do {                                                           \
    A = fmaxf(fmaxf(A, w[(B) + 0] + f0), w[(B) + 1] + f1);       \
    A = fmaxf(fmaxf(A, w[(B) + 2] + f2), w[(B) + 3] + f3);       \
    A = fmaxf(fmaxf(A, w[(B) + 4] + f4), w[(B) + 5] + f5);       \
    A = fmaxf(A, w[(B) + 6] + f6);                               \
  } while (0)

#pragma unroll
  for (int r = 0; r < 10; ++r) {     // input rows oy0 .. oy0+9 (incl. halo)
    const float* rp = &tile[(oy0 + r) * LDSW + ox0];
    const float4 v0 = *(const float4*)(rp);      // cols 0..3  (ds_load_b128)
    const float4 v1 = *(const float4*)(rp + 4);  // cols 4..7  (ds_load_b128)
    const float2 v2 = *(const float2*)(rp + 8);  // cols 8..9  (ds_load_b64)
    const float w[10] = {v0.x, v0.y, v0.z, v0.w,
                         v1.x, v1.y, v1.z, v1.w,
                         v2.x, v2.y};
#pragma unroll
    for (int oy = 0; oy < 4; ++oy) {
      const int i = r - oy;          // filter row feeding output row oy
      if (i < 0 || i >= KX) continue; // folds at compile time (r, oy unrolled)
      const float f0 = fv[i * KX + 0], f1 = fv[i * KX + 1], f2 = fv[i * KX + 2];
      const float f3 = fv[i * KX + 3], f4 = fv[i * KX + 4], f5 = fv[i * KX + 5];
      const float f6 = fv[i * KX + 6];
      APPLY_ROW(acc[oy][0], 0);
      APPLY_ROW(acc[oy][1], 1);
      APPLY_ROW(acc[oy][2], 2);
      APPLY_ROW(acc[oy][3], 3);
    }
  }
#undef APPLY_ROW

#pragma unroll
  for (int oy = 0; oy < 4; ++oy) {
    float4 r4;
    r4.x = acc[oy][0]; r4.y = acc[oy][1]; r4.z = acc[oy][2]; r4.w = acc[oy][3];
    *(float4*)&plane_out[(size_t)(by + oy0 + oy) * WIDTH + bx + ox0] = r4;
  }
}

extern "C" void kernel_launch(void* const* d_in, const int* in_sizes, int n_in,
                              void* d_out, int out_size, void* d_ws, size_t ws_size,
                              hipStream_t stream) {
  (void)in_sizes; (void)n_in; (void)out_size; (void)d_ws; (void)ws_size;
  const float* img  = (const float*)d_in[0];   // (8,16,512,512) fp32
  const float* filt = (const float*)d_in[1];   // (7,7) fp32
  float* out = (float*)d_out;

  dim3 grid(WIDTH / TILE, WIDTH / TILE, PLANES);  // 8 x 8 x 128
  dim3 block(256);
  dilate7x7_kernel<<<grid, block, 0, stream>>>(img, filt, out);
}